// ChamferDistance_9534827397593
// MI455X (gfx1250) — compile-verified
//
#include <hip/hip_runtime.h>
#include <hip/hip_bf16.h>

typedef __attribute__((ext_vector_type(2))) float v2f;
typedef __attribute__((ext_vector_type(8))) float v8f;

#define B_  64
#define NV  1024
#define NL  512
#define DD  512
#define TV  64     // v rows per workgroup
#define TL  128    // l cols per l-iteration
#define KC  64     // K chunk staged in LDS
#define NCHUNK (DD / KC)
#define LSTR 68    // LDS row stride (floats): bank-conflict-free b64 operand reads

// ---------------- norms: nx[b*NV+v] = sum_d x^2 ; ny[b*NL+l] = sum_d y^2 ----
__global__ __launch_bounds__(256) void cd_norms(const float* __restrict__ video,
                                                const float* __restrict__ lang,
                                                float* __restrict__ nx,
                                                float* __restrict__ ny) {
  int rid  = blockIdx.x * 8 + (threadIdx.x >> 5);   // one wave per 512-float row
  int lane = threadIdx.x & 31;
  const float* src;
  float* dst;
  if (rid < B_ * NV) { src = video + (size_t)rid * DD;              dst = nx + rid; }
  else               { int r = rid - B_ * NV;
                       src = lang + (size_t)r * DD;                 dst = ny + r;  }
  float s = 0.f;
  const float4* s4 = (const float4*)src;
  float4 q[4];
#pragma unroll
  for (int p = 0; p < 4; ++p) q[p] = s4[lane + 32 * p];
#pragma unroll
  for (int p = 0; p < 4; ++p)
    s += q[p].x * q[p].x + q[p].y * q[p].y + q[p].z * q[p].z + q[p].w * q[p].w;
#pragma unroll
  for (int m = 16; m >= 1; m >>= 1) s += __shfl_xor(s, m, 32);
  if (lane == 0) *dst = s;
}

// ---------------- init workspace --------------------------------------------
__global__ __launch_bounds__(256) void cd_init(float* __restrict__ colmin,
                                               float* __restrict__ rowsum) {
  int i = blockIdx.x * 256 + threadIdx.x;
  if (i < B_ * NL) colmin[i] = __builtin_inff();
  if (i < B_)      rowsum[i] = 0.f;
}

// ---------------- main tiled WMMA kernel ------------------------------------
__global__ __launch_bounds__(256) void cd_main(const float* __restrict__ video,
                                               const float* __restrict__ lang,
                                               const float* __restrict__ nx,
                                               const float* __restrict__ ny,
                                               float* __restrict__ colmin,
                                               float* __restrict__ rowsum) {
  __shared__ float Xs[2][TV * LSTR];    // 2 x 17.0 KB (ping-pong)
  __shared__ float Ys[2][TL * LSTR];    // 2 x 34.0 KB (ping-pong)
  __shared__ float colminLDS[NL];       //  2.0 KB
  __shared__ float rowminLDS[TV];       //  0.25 KB

  const int tid = threadIdx.x;
  const int b   = blockIdx.y;
  const int v0  = blockIdx.x * TV;

  for (int i = tid; i < NL; i += 256) colminLDS[i] = __builtin_inff();
  if (tid < TV) rowminLDS[tid] = __builtin_inff();

  const int wave = tid >> 5;
  const int lane = tid & 31;
  const int wv   = wave & 3;    // v sub-tile (16 rows each)
  const int wl   = wave >> 2;   // l sub-range (64 cols each)
  const int half = lane >> 4;   // WMMA lane half: M offset +8
  const int n    = lane & 15;   // WMMA N index

  const float* Xg = video + (size_t)b * NV * DD;
  const float* Yg = lang  + (size_t)b * NL * DD;

  // hoisted ||x||^2 for this wave's 16 rows; per lane: m = j + 8*half
  float nxv[8];
#pragma unroll
  for (int j = 0; j < 8; ++j)
    nxv[j] = nx[(size_t)b * NV + v0 + 16 * wv + j + 8 * half];

  float rowminP[8];
#pragma unroll
  for (int j = 0; j < 8; ++j) rowminP[j] = __builtin_inff();

  // WMMA operand LDS offsets (A: 16x4 f32, B: 4x16 f32), k added in loop.
  // lanes 0-15: K={k,k+1}; lanes 16-31: K={k+2,k+3}  -> one b64 per operand.
  const int aBase  = (16 * wv + n) * LSTR + 2 * half;
  const int bBase0 = (64 * wl + n) * LSTR + 2 * half;

  const int lrow = tid >> 4;         // 0..15 (global->LDS staging)
  const int lcol = (tid & 15) * 4;

  for (int l0 = 0; l0 < NL; l0 += TL) {
    v8f acc[4] = {};   // 4 C tiles: l = l0 + 64*wl + 16*t + n
    float4 xr[4], yr[8];

    // ---- prologue: chunk 0 -> regs -> buf 0 (loads pipeline, one wait) ----
#pragma unroll
    for (int p = 0; p < 4; ++p)
      xr[p] = *(const float4*)(Xg + (size_t)(v0 + lrow + 16 * p) * DD + lcol);
#pragma unroll
    for (int p = 0; p < 8; ++p)
      yr[p] = *(const float4*)(Yg + (size_t)(l0 + lrow + 16 * p) * DD + lcol);
#pragma unroll
    for (int p = 0; p < 4; ++p)
      *(float4*)&Xs[0][(lrow + 16 * p) * LSTR + lcol] = xr[p];
#pragma unroll
    for (int p = 0; p < 8; ++p)
      *(float4*)&Ys[0][(lrow + 16 * p) * LSTR + lcol] = yr[p];

    for (int c = 0; c < NCHUNK; ++c) {
      __syncthreads();                 // buf[c&1] ready for everyone
      const int cur = c & 1;
      const int nxt = cur ^ 1;
      const bool pf = (c + 1) < NCHUNK;
      const int kn  = (c + 1) * KC;

      // issue next chunk's global loads BEFORE compute (latency hidden)
      if (pf) {
#pragma unroll
        for (int p = 0; p < 4; ++p)
          xr[p] = *(const float4*)(Xg + (size_t)(v0 + lrow + 16 * p) * DD + kn + lcol);
#pragma unroll
        for (int p = 0; p < 8; ++p)
          yr[p] = *(const float4*)(Yg + (size_t)(l0 + lrow + 16 * p) * DD + kn + lcol);
      }

      // ---- 64 WMMAs on current buffer ----
      const float* xb = Xs[cur];
      const float* yb = Ys[cur];
#pragma unroll
      for (int k = 0; k < KC; k += 4) {
        v2f a = *(const v2f*)&xb[aBase + k];
#pragma unroll
        for (int t = 0; t < 4; ++t) {
          v2f bb = *(const v2f*)&yb[bBase0 + 16 * t * LSTR + k];
          acc[t] = __builtin_amdgcn_wmma_f32_16x16x4_f32(
              false, a, false, bb, (short)0, acc[t], false, false);
        }
      }

      // stash prefetched chunk into the other buffer
      if (pf) {
#pragma unroll
        for (int p = 0; p < 4; ++p)
          *(float4*)&Xs[nxt][(lrow + 16 * p) * LSTR + lcol] = xr[p];
#pragma unroll
        for (int p = 0; p < 8; ++p)
          *(float4*)&Ys[nxt][(lrow + 16 * p) * LSTR + lcol] = yr[p];
      }
    }

    // epilogue for this l-range: dist = ||x||^2 + ||y||^2 - 2*ab
#pragma unroll
    for (int t = 0; t < 4; ++t) {
      int l = l0 + 64 * wl + 16 * t + n;
      float nyl  = ny[(size_t)b * NL + l];
      float cmin = __builtin_inff();
#pragma unroll
      for (int j = 0; j < 8; ++j) {
        float d = nxv[j] + nyl - 2.0f * acc[t][j];
        rowminP[j] = fminf(rowminP[j], d);
        cmin       = fminf(cmin, d);
      }
      // min over all 16 M rows of the wave (combine lane halves)
      cmin = fminf(cmin, __shfl_xor(cmin, 16, 32));
      if (half == 0) atomicMin(&colminLDS[l], cmin);   // ds_min_num_f32
    }
  }

  // row mins: reduce over N (16 lanes within each half), fold into LDS
#pragma unroll
  for (int j = 0; j < 8; ++j) {
    float r = rowminP[j];
#pragma unroll
    for (int m = 8; m >= 1; m >>= 1) r = fminf(r, __shfl_xor(r, m, 32));
    rowminP[j] = r;
  }
  if (n == 0) {  // lanes 0 and 16: m = j and j+8
#pragma unroll
    for (int j = 0; j < 8; ++j)
      atomicMin(&rowminLDS[16 * wv + j + 8 * half], rowminP[j]);
  }
  __syncthreads();

  // sum the 64 row-mins -> one global atomicAdd per workgroup
  if (tid < 32) {
    float s = rowminLDS[tid] + rowminLDS[tid + 32];
#pragma unroll
    for (int m = 16; m >= 1; m >>= 1) s += __shfl_xor(s, m, 32);
    if (tid == 0) atomicAdd(&rowsum[b], s);
  }
  // column mins -> global atomic min (combined across the 16 v-tiles)
  for (int i = tid; i < NL; i += 256)
    atomicMin(&colmin[(size_t)b * NL + i], colminLDS[i]);
}

// ---------------- final: out[b] = rowsum/Nv + sum(colmin)/Nl ----------------
__global__ __launch_bounds__(256) void cd_final(const float* __restrict__ colmin,
                                                const float* __restrict__ rowsum,
                                                float* __restrict__ out) {
  __shared__ float red[256];
  int b = blockIdx.x, tid = threadIdx.x;
  float s = 0.f;
  for (int i = tid; i < NL; i += 256) s += colmin[(size_t)b * NL + i];
  red[tid] = s;
  __syncthreads();
  for (int off = 128; off >= 32; off >>= 1) {
    if (tid < off) red[tid] += red[tid + off];
    __syncthreads();
  }
  if (tid < 32) {
    float v = red[tid];
#pragma unroll
    for (int m = 16; m >= 1; m >>= 1) v += __shfl_xor(v, m, 32);
    if (tid == 0) out[b] = rowsum[b] * (1.0f / NV) + v * (1.0f / NL);
  }
}

extern "C" void kernel_launch(void* const* d_in, const int* in_sizes, int n_in,
                              void* d_out, int out_size, void* d_ws, size_t ws_size,
                              hipStream_t stream) {
  const float* video = (const float*)d_in[0];   // [64,1024,512] f32
  const float* lang  = (const float*)d_in[1];   // [64, 512,512] f32
  float* out = (float*)d_out;                   // [64] f32

  float* ws     = (float*)d_ws;
  float* nx     = ws;                 // 65536 floats
  float* ny     = nx + B_ * NV;       // 32768 floats
  float* colmin = ny + B_ * NL;       // 32768 floats
  float* rowsum = colmin + B_ * NL;   // 64 floats   (total ~513 KB)

  cd_init <<<(B_ * NL + 255) / 256, 256, 0, stream>>>(colmin, rowsum);
  cd_norms<<<(B_ * NV + B_ * NL) / 8, 256, 0, stream>>>(video, lang, nx, ny);
  cd_main <<<dim3(NV / TV, B_), 256, 0, stream>>>(video, lang, nx, ny, colmin, rowsum);
  cd_final<<<B_, 256, 0, stream>>>(colmin, rowsum, out);
}